// VectorExpansion_23450521436918
// MI455X (gfx1250) — compile-verified
//
#include <hip/hip_runtime.h>
#include <hip/hip_bf16.h>
#include <stdint.h>

// VectorExpansion (LE radial basis) for MI455X / gfx1250.
// Bandwidth-bound: ~256MB out + ~100MB in => ~16us floor at 23.3 TB/s.
// CDNA5 paths used:
//   * TDM tensor_load_to_lds: one DMA descriptor stages the cells table to LDS
//     (TENSORcnt + s_wait_tensorcnt).
//   * global_load_async_to_lds_b32/b64: double-buffered ASYNCcnt pipeline that
//     stages the streaming index arrays (pairs / cell_shifts / structure_pairs)
//     into LDS one tile ahead of compute (s_wait_asynccnt 5).
// Compute: 1 v_sin + 1 v_cos per pair + Chebyshev recurrence for sin(n*pi*x);
// output written as two b128 stores per l (lane-coalesced 32B chunks).

#define RB_N_MAX 8
#define RB_L_MAX 3
#define TILE 256
#define TPB  4                       // tiles per block (double-buffered)

static constexpr int S_CAP = 64;     // reference uses S = 32 structures

typedef uint32_t u32x4 __attribute__((ext_vector_type(4)));
typedef uint32_t u32x8 __attribute__((ext_vector_type(8)));

__device__ __forceinline__ uint32_t lds_lo32(const void* p) {
    // Generic (flat) LDS pointers map to LDS via addr[31:0] truncation.
    return (uint32_t)(uintptr_t)p;
}

__device__ __forceinline__ void async_b32(uint32_t lds, const void* base, int off) {
    asm volatile("global_load_async_to_lds_b32 %0, %1, %2"
                 :: "v"(lds), "v"(off), "s"(base) : "memory");
}
__device__ __forceinline__ void async_b64(uint32_t lds, const void* base, int off) {
    asm volatile("global_load_async_to_lds_b64 %0, %1, %2"
                 :: "v"(lds), "v"(off), "s"(base) : "memory");
}
template <int N>
__device__ __forceinline__ void wait_async() {
    asm volatile("s_wait_asynccnt %0" :: "n"(N) : "memory");
}
__device__ __forceinline__ void wait_ds0() {
    asm volatile("s_wait_dscnt 0x0" ::: "memory");
}

// 1-D TDM copy: nelem 4-byte elements from gptr -> LDS at lds_addr.
// D# per CDNA5 ISA ch.8: group0 {count=1, lds_addr, global_addr, type=2},
// group1 {data_size=4B, tensor_dim0=nelem, tensor_dim1=1, tile_dim0=nelem,
//         tile_dim1=1, dim0_stride=nelem}.
__device__ __forceinline__ void tdm_load_1d(uint32_t lds_addr, const void* gptr,
                                            uint32_t nelem) {
    const uint64_t ga = (uint64_t)(uintptr_t)gptr;
    u32x4 g0;
    g0.x = 1u;                                             // count=1, user mode
    g0.y = lds_addr;                                       // lds_addr [63:32]
    g0.z = (uint32_t)ga;                                   // global_addr lo
    g0.w = (uint32_t)((ga >> 32) & 0x01FFFFFFu) | (2u << 30); // addr hi | type=2
    u32x8 g1;
    g1.s0 = (2u << 16);                                    // data_size=2 (4B)
    g1.s1 = (nelem & 0xFFFFu) << 16;                       // tensor_dim0 lo16
    g1.s2 = (nelem >> 16) | (1u << 16);                    // dim0 hi16 | tensor_dim1=1
    g1.s3 = (nelem & 0xFFFFu) << 16;                       // dim1 hi=0 | tile_dim0
    g1.s4 = 1u;                                            // tile_dim1=1, tile_dim2=0
    g1.s5 = nelem;                                         // tensor_dim0_stride lo32
    g1.s6 = 0u;                                            // stride hi | dim1_stride lo
    g1.s7 = 0u;
    asm volatile("tensor_load_to_lds %0, %1" :: "s"(g0), "s"(g1) : "memory");
}

__global__ __launch_bounds__(256)
void VectorExpansion_kernel(
    const float* __restrict__ positions,          // [N,3]
    const float* __restrict__ cells,              // [S,3,3]
    const int*   __restrict__ cell_shifts,        // [P,3]
    const int*   __restrict__ pairs,              // [P,2]
    const int*   __restrict__ structure_pairs,    // [P]
    const int*   __restrict__ structure_offsets,  // [S]
    float* __restrict__ out,                      // [L_MAX+1, P, N_MAX]
    int P, int S)
{
    __shared__ alignas(16) float sCell[S_CAP * 9];
    __shared__ int   sOff[S_CAP];
    __shared__ alignas(16) int sPair[2][TILE * 2];   // per-tile [i0,j0]
    __shared__ alignas(16) int sShift[2][TILE * 3];  // per-tile shifts
    __shared__ alignas(16) int sSp[2][TILE];         // per-tile structure ids

    const int t = threadIdx.x;

    // --- TDM: wave 0 DMAs the cells table into LDS (one descriptor) ---
    if (t < 32) {
        tdm_load_1d(lds_lo32(sCell), cells, (uint32_t)(S * 9));
        __builtin_amdgcn_s_wait_tensorcnt(0);
    }
    if (t < S) sOff[t] = structure_offsets[t];
    __syncthreads();

    const int64_t base0 = (int64_t)blockIdx.x * (TILE * TPB);

    // Stage one tile of index data into LDS via async (ASYNCcnt) loads.
    auto issue = [&](int k, int buf) {
        int64_t p64 = base0 + (int64_t)k * TILE + t;
        int p = (int)((p64 < P) ? p64 : (int64_t)(P - 1));   // clamp tail
        async_b64(lds_lo32(&sPair[buf][2 * t]),      pairs,           p * 8);
        async_b32(lds_lo32(&sShift[buf][3 * t + 0]), cell_shifts,     p * 12 + 0);
        async_b32(lds_lo32(&sShift[buf][3 * t + 1]), cell_shifts,     p * 12 + 4);
        async_b32(lds_lo32(&sShift[buf][3 * t + 2]), cell_shifts,     p * 12 + 8);
        async_b32(lds_lo32(&sSp[buf][t]),            structure_pairs, p * 4);
    };

    issue(0, 0);

    const size_t plane = (size_t)P * RB_N_MAX;

#pragma unroll
    for (int k = 0; k < TPB; ++k) {
        const int buf = k & 1;
        if (k + 1 < TPB) {
            issue(k + 1, (k + 1) & 1);   // fire next tile first, then
            wait_async<5>();             // wait for this tile's 5 loads (in-order)
        } else {
            wait_async<0>();
        }

        const int64_t p64 = base0 + (int64_t)k * TILE + t;
        if (p64 < P) {
            const int p = (int)p64;
            // ---- read staged indices from LDS (lane-private) ----
            const int i0 = sPair[buf][2 * t + 0];
            const int j0 = sPair[buf][2 * t + 1];
            const int sp = sSp[buf][t];
            const float sx = (float)sShift[buf][3 * t + 0];
            const float sy = (float)sShift[buf][3 * t + 1];
            const float sz = (float)sShift[buf][3 * t + 2];

            const int off = sOff[sp];
            const int ia  = off + i0;
            const int ja  = off + j0;
            const float* __restrict__ c = &sCell[sp * 9];

            const float vx = positions[3 * ja + 0] - positions[3 * ia + 0]
                           + sx * c[0] + sy * c[3] + sz * c[6];
            const float vy = positions[3 * ja + 1] - positions[3 * ia + 1]
                           + sx * c[1] + sy * c[4] + sz * c[7];
            const float vz = positions[3 * ja + 2] - positions[3 * ia + 2]
                           + sx * c[2] + sy * c[5] + sz * c[8];

            const float r2 = vx * vx + vy * vy + vz * vz + 1e-12f;
            const float r  = __builtin_sqrtf(r2);
            const float inv_r = __frcp_rn(r);        // r >= 1e-6 > EPS
            const float x  = r * 0.2f;               // r / R_CUT
            const float th = 3.14159265358979323846f * x;
            const float s1 = __sinf(th);
            const float c1 = __cosf(th);
            const float fc = (x < 1.0f) ? 0.5f * (c1 + 1.0f) : 0.0f;
            const float scale = fc * 0.6324555320336759f * inv_r;

            float base[RB_N_MAX];
            {
                const float twoc = 2.0f * c1;
                float sm2 = 0.0f, sm1 = s1;
                base[0] = s1 * scale;
#pragma unroll
                for (int n = 1; n < RB_N_MAX; ++n) {
                    const float sn = __builtin_fmaf(twoc, sm1, -sm2);
                    sm2 = sm1; sm1 = sn;
                    base[n] = sn * scale;
                }
            }

            float xl = 1.0f;
#pragma unroll
            for (int l = 0; l <= RB_L_MAX; ++l) {
                float4 a, b;
                a.x = base[0] * xl; a.y = base[1] * xl; a.z = base[2] * xl; a.w = base[3] * xl;
                b.x = base[4] * xl; b.y = base[5] * xl; b.z = base[6] * xl; b.w = base[7] * xl;
                float* dst = out + (size_t)l * plane + (size_t)p * RB_N_MAX;
                *reinterpret_cast<float4*>(dst)     = a;
                *reinterpret_cast<float4*>(dst + 4) = b;
                xl *= x;
            }
        }
        wait_ds0();   // drain this tile's LDS reads before its buffer is re-armed
    }
}

extern "C" void kernel_launch(void* const* d_in, const int* in_sizes, int n_in,
                              void* d_out, int out_size, void* d_ws, size_t ws_size,
                              hipStream_t stream) {
    // setup_inputs() order:
    // 0 positions f32 | 1 cells f32 | 2 species i32 | 3 cell_shifts i32
    // 4 centers i32 | 5 pairs i32 | 6 structure_centers i32
    // 7 structure_pairs i32 | 8 structure_offsets i32
    const float* positions         = (const float*)d_in[0];
    const float* cells             = (const float*)d_in[1];
    const int*   cell_shifts       = (const int*)d_in[3];
    const int*   pairs             = (const int*)d_in[5];
    const int*   structure_pairs   = (const int*)d_in[7];
    const int*   structure_offsets = (const int*)d_in[8];
    float* out = (float*)d_out;

    const int P = in_sizes[7];          // total pairs
    const int S = in_sizes[8];          // structures (<= S_CAP)

    const int perBlock = TILE * TPB;
    const int grid = (P + perBlock - 1) / perBlock;
    VectorExpansion_kernel<<<grid, TILE, 0, stream>>>(
        positions, cells, cell_shifts, pairs, structure_pairs, structure_offsets,
        out, P, S);
}